// GCNNet1_82145544503550
// MI455X (gfx1250) — compile-verified
//
#include <hip/hip_runtime.h>
#include <math.h>

// ---------------------------------------------------------------------------
// GCN (3x GCNConv + PReLU + global max pool) for MI455X / gfx1250.
// Dense transforms run on V_WMMA_F32_16X16X4_F32 (native fp32 WMMA, exact
// precision vs the fp32 reference). Each 128-thread block stages a 64xK fp32
// A-slab into LDS once (shared by its 4 waves = 4 adjacent N-tiles, cutting
// global A traffic 4x); waves read A fragments via ds_load_b64 (low latency)
// while the pre-swizzled B fragments stream from global as one coalesced
// global_load_b64 per K-step. Aggregation is scatter-atomic and L2-resident.
// ---------------------------------------------------------------------------

typedef __attribute__((ext_vector_type(2))) float v2f;
typedef __attribute__((ext_vector_type(8))) float v8f;

#define CF    108
#define GBAT  128
#define COUT3 432
#define LDSTRIDE_MAX 218   // K_max(216) + 2 pad (even, bank-friendly)

__device__ __forceinline__ v8f wmma_f32_16x16x4(v2f a, v2f b, v8f c) {
  // 8 args: (neg_a, A, neg_b, B, c_mod, C, reuse_a, reuse_b)
  return __builtin_amdgcn_wmma_f32_16x16x4_f32(false, a, false, b, (short)0, c,
                                               false, false);
}

// -------------------- degree / symmetric normalization ---------------------
__global__ void k_deg_init(float* deg, int n) {
  int i = blockIdx.x * blockDim.x + threadIdx.x;
  if (i < n) deg[i] = 1.0f;  // self loop
}
__global__ void k_deg_edges(const int* __restrict__ col, float* deg, int e) {
  int i = blockIdx.x * blockDim.x + threadIdx.x;
  if (i < e) atomicAdd(&deg[col[i]], 1.0f);
}
__global__ void k_rsqrt_inplace(float* deg, int n) {
  int i = blockIdx.x * blockDim.x + threadIdx.x;
  if (i < n) deg[i] = rsqrtf(deg[i]);
}

// -------------------- weight swizzle into WMMA B-fragment order ------------
// Wswz[nT][ks][lane*2+j] = W[4*ks + 2*(lane>=16) + j][16*nT + (lane&15)]
// (zero-padded past Cout). One K-step's B fragment for a wave is then a
// single contiguous 256B block -> one global_load_b64 per lane.
__global__ void k_swz_w(const float* __restrict__ W, float* __restrict__ Wswz,
                        int Cin, int Cout, int Cpad) {
  int t = blockIdx.x * blockDim.x + threadIdx.x;
  int Ksteps = Cin >> 2;
  int nTiles = Cpad >> 4;
  if (t >= nTiles * Ksteps * 64) return;
  int nT   = t / (Ksteps << 6);
  int rem  = t - nT * (Ksteps << 6);
  int ks   = rem >> 6;
  int idx  = rem & 63;
  int lane = idx >> 1;
  int j    = idx & 1;
  int k = (ks << 2) + ((lane >> 4) << 1) + j;
  int c = (nT << 4) + (lane & 15);
  Wswz[t] = (c < Cout) ? W[k * Cout + c] : 0.0f;
}
__global__ void k_pad_b(const float* __restrict__ b, float* __restrict__ bp,
                        int Cout, int Cpad) {
  int c = blockIdx.x * blockDim.x + threadIdx.x;
  if (c < Cpad) bp[c] = (c < Cout) ? b[c] : 0.0f;
}

// -------------------- dense transform: T[N,Cpad] = A[N,K] @ W[K,Cpad] ------
// Block = 4 waves sharing one 64-row slab (staged in LDS), covering 4
// adjacent 16-col N-tiles. Each wave: 4 stacked 16x16 WMMA accumulators
// sharing a single pre-swizzled B fragment per K-step (K stepped by 4).
__global__ void __launch_bounds__(128) k_gemm_wmma(
    const float* __restrict__ A, const float* __restrict__ Wswz,
    float* __restrict__ T, int nRows, int K, int ldA, int Cpad) {
  __shared__ float lds[64 * LDSTRIDE_MAX];

  const int tid    = threadIdx.x;
  const int lane   = tid & 31;
  const int waveId = tid >> 5;
  const int nTiles  = Cpad >> 4;
  const int nGroups = (nTiles + 3) >> 2;
  const int bid     = blockIdx.x;
  const int mSuper  = bid / nGroups;
  const int g       = bid - mSuper * nGroups;
  const int rowBase = mSuper << 6;
  const int nT      = (g << 2) + waveId;
  const int ldL     = K + 2;  // even -> 8B-aligned b64; odd/32 -> few conflicts

  // ---- cooperative fill: 64 rows x K floats of A into LDS ----
  const int halfK = K >> 1;
  for (int idx = tid; idx < (halfK << 6); idx += 128) {
    int rrow = idx / halfK;
    int c2   = idx - rrow * halfK;
    int gr = rowBase + rrow;
    if (gr >= nRows) gr = nRows - 1;  // clamp: duplicates are never stored
    v2f v = *(const v2f*)(A + (size_t)gr * ldA + (c2 << 1));
    *(v2f*)(&lds[rrow * ldL + (c2 << 1)]) = v;
  }
  __syncthreads();

  if (nT < nTiles) {  // wave-uniform guard: EXEC stays all-1s for WMMA
    const int m   = lane & 15;        // A row within tile / B column
    const int hi  = lane >> 4;        // lane half
    const int khi = hi << 1;          // K offset 0 or 2 (32-bit A/B layout)
    const int Ksteps = K >> 2;

    const float* l0 = &lds[m * ldL + khi];
    const float* l1 = l0 + (ldL << 4);  // +16 rows
    const float* l2 = l0 + (ldL << 5);  // +32 rows
    const float* l3 = l0 + 48 * ldL;    // +48 rows
    const float* wp = Wswz + (((size_t)nT * Ksteps) << 6) + (lane << 1);

    v8f acc0 = {}, acc1 = {}, acc2 = {}, acc3 = {};
    for (int ks = 0; ks < Ksteps; ++ks) {
      v2f b = *(const v2f*)wp;  // one coalesced b64: full B fragment
      wp += 64;
      v2f f0 = *(const v2f*)l0; l0 += 4;
      v2f f1 = *(const v2f*)l1; l1 += 4;
      v2f f2 = *(const v2f*)l2; l2 += 4;
      v2f f3 = *(const v2f*)l3; l3 += 4;
      acc0 = wmma_f32_16x16x4(f0, b, acc0);
      acc1 = wmma_f32_16x16x4(f1, b, acc1);
      acc2 = wmma_f32_16x16x4(f2, b, acc2);
      acc3 = wmma_f32_16x16x4(f3, b, acc3);
    }

    // C/D layout: N = lane%16, M = vgpr + 8*(lane>=16)
    const int nOff = (nT << 4) + m;
    const int mo = hi << 3;
#pragma unroll
    for (int j = 0; j < 8; ++j) {
      int rr = rowBase + mo + j;
      if (rr < nRows) T[(size_t)rr * Cpad + nOff] = acc0[j];
    }
#pragma unroll
    for (int j = 0; j < 8; ++j) {
      int rr = rowBase + 16 + mo + j;
      if (rr < nRows) T[(size_t)rr * Cpad + nOff] = acc1[j];
    }
#pragma unroll
    for (int j = 0; j < 8; ++j) {
      int rr = rowBase + 32 + mo + j;
      if (rr < nRows) T[(size_t)rr * Cpad + nOff] = acc2[j];
    }
#pragma unroll
    for (int j = 0; j < 8; ++j) {
      int rr = rowBase + 48 + mo + j;
      if (rr < nRows) T[(size_t)rr * Cpad + nOff] = acc3[j];
    }
  }
}

// -------------------- aggregation ------------------------------------------
// A_out[i,:] = dinv[i]^2 * T[i,:] + b  (self loop + bias, full overwrite)
__global__ void k_agg_init(const float* __restrict__ T,
                           const float* __restrict__ dinv,
                           const float* __restrict__ bp,
                           float* __restrict__ Aout, int n, int Cpad) {
  int t = blockIdx.x * blockDim.x + threadIdx.x;
  int perRow = Cpad >> 2;
  if (t >= n * perRow) return;
  int i = t / perRow, c4 = t - i * perRow;
  float d = dinv[i];
  float s = d * d;
  float4 tv = ((const float4*)(T + (size_t)i * Cpad))[c4];
  float4 bv = ((const float4*)bp)[c4];
  float4 o;
  o.x = tv.x * s + bv.x;
  o.y = tv.y * s + bv.y;
  o.z = tv.z * s + bv.z;
  o.w = tv.w * s + bv.w;
  ((float4*)(Aout + (size_t)i * Cpad))[c4] = o;
}

// A_out[col,:] += dinv[row]*dinv[col] * T[row,:]   (scatter-atomic)
__global__ void k_agg_edges(const int* __restrict__ row,
                            const int* __restrict__ col,
                            const float* __restrict__ dinv,
                            const float* __restrict__ T,
                            float* __restrict__ Aout, int e, int Cpad) {
  int t = blockIdx.x * blockDim.x + threadIdx.x;
  int perRow = Cpad >> 2;
  if (t >= e * perRow) return;
  int ei = t / perRow, c4 = t - ei * perRow;
  int r = row[ei], c = col[ei];
  float w = dinv[r] * dinv[c];
  float4 tv = ((const float4*)(T + (size_t)r * Cpad))[c4];
  float* dst = Aout + (size_t)c * Cpad + (c4 << 2);
  atomicAdd(dst + 0, tv.x * w);
  atomicAdd(dst + 1, tv.y * w);
  atomicAdd(dst + 2, tv.z * w);
  atomicAdd(dst + 3, tv.w * w);
}

__global__ void k_prelu(float* __restrict__ A, const float* __restrict__ pa,
                        int total) {
  int t = blockIdx.x * blockDim.x + threadIdx.x;
  if (t >= total) return;
  float a = pa[0];
  float v = A[t];
  A[t] = (v >= 0.0f) ? v : a * v;
}

// -------------------- output: -inf init + fused prelu/max-pool -------------
__global__ void k_out_init(float* out, int total) {
  int t = blockIdx.x * blockDim.x + threadIdx.x;
  if (t < total) out[t] = __uint_as_float(0xFF800000u);  // -inf
}

__global__ void k_prelu_maxpool(const float* __restrict__ A,
                                const float* __restrict__ pa,
                                const int* __restrict__ batch,
                                float* __restrict__ out, int n, int Cp) {
  int t = blockIdx.x * blockDim.x + threadIdx.x;
  if (t >= n * Cp) return;
  int i = t / Cp, c = t - i * Cp;
  float a = pa[0];
  float v = A[(size_t)i * Cp + c];
  v = (v >= 0.0f) ? v : a * v;
  float* dst = &out[(size_t)batch[i] * Cp + c];
  // ordered-int float max (works against -inf init, mixed signs)
  if (v >= 0.0f)
    atomicMax((int*)dst, __float_as_int(v));
  else
    atomicMin((unsigned int*)dst, __float_as_uint(v));
}

// ---------------------------------------------------------------------------
static inline unsigned nblk(long long total, int bs) {
  return (unsigned)((total + bs - 1) / bs);
}

extern "C" void kernel_launch(void* const* d_in, const int* in_sizes, int n_in,
                              void* d_out, int out_size, void* d_ws,
                              size_t ws_size, hipStream_t stream) {
  const float* x     = (const float*)d_in[0];
  const int*   ei    = (const int*)d_in[1];
  const int*   batch = (const int*)d_in[2];
  const float* W1    = (const float*)d_in[3];
  const float* b1    = (const float*)d_in[4];
  const float* W2    = (const float*)d_in[5];
  const float* b2    = (const float*)d_in[6];
  const float* W3    = (const float*)d_in[7];
  const float* b3    = (const float*)d_in[8];
  const float* pa    = (const float*)d_in[9];
  float*       out   = (float*)d_out;

  const int N = in_sizes[0] / CF;  // 50000
  const int E = in_sizes[1] / 2;   // 400000
  const int* row = ei;             // edge_index[0] = source
  const int* col = ei + E;         // edge_index[1] = target

  // workspace carve-up (floats)
  float* ws   = (float*)d_ws;
  float* dinv = ws;                                       // N
  float* Wp   = ws + (((size_t)N + 255) & ~(size_t)255);  // up to 216*432
  float* bp   = Wp + 93440;                               // up to 432
  float* B0   = bp + 512;                                 // N * 432 ping
  float* B1   = B0 + (size_t)N * COUT3;                   // N * 432 pong

  // --- normalization coefficients + output init ---
  k_deg_init<<<nblk(N, 256), 256, 0, stream>>>(dinv, N);
  k_deg_edges<<<nblk(E, 256), 256, 0, stream>>>(col, dinv, E);
  k_rsqrt_inplace<<<nblk(N, 256), 256, 0, stream>>>(dinv, N);
  k_out_init<<<nblk((long long)GBAT * COUT3, 256), 256, 0, stream>>>(
      out, GBAT * COUT3);

  const int mSupers = (N + 63) >> 6;

  // --- layer 1: 108 -> 108 (pad 112) ---
  {
    const int Cin = CF, Cout = CF, Cpad = 112;
    k_swz_w<<<nblk((long long)Cin * Cpad, 256), 256, 0, stream>>>(W1, Wp, Cin,
                                                                  Cout, Cpad);
    k_pad_b<<<nblk(Cpad, 256), 256, 0, stream>>>(b1, bp, Cout, Cpad);
    int nGroups = ((Cpad >> 4) + 3) >> 2;
    k_gemm_wmma<<<mSupers * nGroups, 128, 0, stream>>>(x, Wp, B0, N, Cin, Cin,
                                                       Cpad);
    k_agg_init<<<nblk((long long)N * (Cpad >> 2), 256), 256, 0, stream>>>(
        B0, dinv, bp, B1, N, Cpad);
    k_agg_edges<<<nblk((long long)E * (Cpad >> 2), 256), 256, 0, stream>>>(
        row, col, dinv, B0, B1, E, Cpad);
    k_prelu<<<nblk((long long)N * Cpad, 256), 256, 0, stream>>>(B1, pa,
                                                                N * Cpad);
  }

  // --- layer 2: 108 -> 216 (pad 224), A stride 112 ---
  {
    const int Cin = CF, ldA = 112, Cout = 2 * CF, Cpad = 224;
    k_swz_w<<<nblk((long long)Cin * Cpad, 256), 256, 0, stream>>>(W2, Wp, Cin,
                                                                  Cout, Cpad);
    k_pad_b<<<nblk(Cpad, 256), 256, 0, stream>>>(b2, bp, Cout, Cpad);
    int nGroups = ((Cpad >> 4) + 3) >> 2;
    k_gemm_wmma<<<mSupers * nGroups, 128, 0, stream>>>(B1, Wp, B0, N, Cin, ldA,
                                                       Cpad);
    k_agg_init<<<nblk((long long)N * (Cpad >> 2), 256), 256, 0, stream>>>(
        B0, dinv, bp, B1, N, Cpad);
    k_agg_edges<<<nblk((long long)E * (Cpad >> 2), 256), 256, 0, stream>>>(
        row, col, dinv, B0, B1, E, Cpad);
    k_prelu<<<nblk((long long)N * Cpad, 256), 256, 0, stream>>>(B1, pa,
                                                                N * Cpad);
  }

  // --- layer 3: 216 -> 432 (exact), A stride 224; fused prelu + max pool ---
  {
    const int Cin = 2 * CF, ldA = 224, Cout = COUT3, Cpad = COUT3;
    k_swz_w<<<nblk((long long)Cin * Cpad, 256), 256, 0, stream>>>(W3, Wp, Cin,
                                                                  Cout, Cpad);
    k_pad_b<<<nblk(Cpad, 256), 256, 0, stream>>>(b3, bp, Cout, Cpad);
    int nGroups = ((Cpad >> 4) + 3) >> 2;
    k_gemm_wmma<<<mSupers * nGroups, 128, 0, stream>>>(B1, Wp, B0, N, Cin, ldA,
                                                       Cpad);
    k_agg_init<<<nblk((long long)N * (Cpad >> 2), 256), 256, 0, stream>>>(
        B0, dinv, bp, B1, N, Cpad);
    k_agg_edges<<<nblk((long long)E * (Cpad >> 2), 256), 256, 0, stream>>>(
        row, col, dinv, B0, B1, E, Cpad);
    k_prelu_maxpool<<<nblk((long long)N * Cpad, 256), 256, 0, stream>>>(
        B1, pa, batch, out, N, Cpad);
  }
}